// Lp_Conv2D_BT_15238543966607
// MI455X (gfx1250) — compile-verified
//
#include <hip/hip_runtime.h>
#include <hip/hip_bf16.h>
#include <cstdint>
#include <cstddef>

// Problem constants (match the reference)
#define TB      32
#define TC_IN   64
#define THW     64
#define TC_OUT  128
#define CONN    4
#define PLANE   (THW * THW)   // 4096 floats per channel plane

typedef int v4i __attribute__((ext_vector_type(4)));

// gfx1250 async global->LDS path (guarded so either toolchain compiles)
#if defined(__HIP_DEVICE_COMPILE__) && defined(__gfx1250__)
#if __has_builtin(__builtin_amdgcn_global_load_async_to_lds_b128)
#define USE_ASYNC_LDS 1
#endif
#endif

__global__ __launch_bounds__(256) void lp_conv_bt_kernel(
    const float* __restrict__ x,      // (B, C_IN, 64, 64)
    const float* __restrict__ w,      // (C_OUT, 4)
    const float* __restrict__ bias,   // (C_OUT)
    const int*   __restrict__ conn,   // (C_OUT, 4), values in [0, 576)
    float*       __restrict__ out)    // (B, C_OUT, 64, 64)
{
    __shared__ __align__(16) float smem[CONN * PLANE];   // 64 KB

    const int o   = blockIdx.x;    // 0..127
    const int b   = blockIdx.y;    // 0..31
    const int tid = threadIdx.x;   // 0..255

    // Block-uniform parameters -> scalar loads
    const int4   ci = *reinterpret_cast<const int4*>(conn + o * CONN);
    const float4 wv = *reinterpret_cast<const float4*>(w + o * CONN);
    const float  bo = bias[o];

    int cc[CONN], di[CONN], dj[CONN];
    {
        const int idx[CONN] = {ci.x, ci.y, ci.z, ci.w};
#pragma unroll
        for (int k = 0; k < CONN; ++k) {
            const int c   = idx[k] / 9;        // input channel
            const int rem = idx[k] - c * 9;
            cc[k] = c;
            di[k] = rem / 3 - 1;               // fold pad_l = 1 into the tap offset
            dj[k] = rem % 3 - 1;
        }
    }
    const float wk[CONN] = {wv.x, wv.y, wv.z, wv.w};

    // ---- Stage the 4 required input planes into LDS ----
#pragma unroll
    for (int k = 0; k < CONN; ++k) {
        const float4* g = reinterpret_cast<const float4*>(
            x + ((size_t)b * TC_IN + cc[k]) * PLANE);
        float4* l = reinterpret_cast<float4*>(smem + k * PLANE);
#pragma unroll
        for (int t = 0; t < PLANE / 4 / 256; ++t) {   // 4 float4 per thread per plane
            const int e = tid + t * 256;
#if defined(USE_ASYNC_LDS)
            __builtin_amdgcn_global_load_async_to_lds_b128(
                (v4i*)(g + e),            // global source
                (v4i*)(l + e),            // LDS destination
                /*offset=*/0, /*cpol=*/0);
#else
            l[e] = g[e];
#endif
        }
    }
#if defined(USE_ASYNC_LDS)
#if __has_builtin(__builtin_amdgcn_s_wait_asynccnt)
    __builtin_amdgcn_s_wait_asynccnt(0);
#else
    asm volatile("s_wait_asynccnt 0" ::: "memory");
#endif
#endif
    __syncthreads();

    // ---- Compute: each thread owns column j, 16 rows ----
    const int j  = tid & (THW - 1);
    const int i0 = (tid >> 6) << 4;

    // Clamped column index per connection (edge padding == clamp)
    int jc[CONN];
#pragma unroll
    for (int k = 0; k < CONN; ++k) jc[k] = min(max(j + dj[k], 0), THW - 1);

    float* orow = out + ((size_t)b * TC_OUT + o) * PLANE;

#pragma unroll 4
    for (int r = 0; r < 16; ++r) {
        const int i = i0 + r;
        float m = 0.0f;   // all candidates are |.| >= 0, so 0 is a safe identity
#pragma unroll
        for (int k = 0; k < CONN; ++k) {
            const int ic  = min(max(i + di[k], 0), THW - 1);
            const float v = smem[k * PLANE + ic * THW + jc[k]];
            m = fmaxf(m, fabsf(wk[k] - v));
        }
        orow[i * THW + j] = m + bo;
    }
}

extern "C" void kernel_launch(void* const* d_in, const int* in_sizes, int n_in,
                              void* d_out, int out_size, void* d_ws, size_t ws_size,
                              hipStream_t stream) {
    (void)in_sizes; (void)n_in; (void)out_size; (void)d_ws; (void)ws_size;
    const float* x    = (const float*)d_in[0];
    const float* w    = (const float*)d_in[1];
    const float* bias = (const float*)d_in[2];
    const int*   conn = (const int*)d_in[3];
    float* out = (float*)d_out;

    dim3 grid(TC_OUT, TB);   // 128 x 32 workgroups
    lp_conv_bt_kernel<<<grid, 256, 0, stream>>>(x, w, bias, conn, out);
}